// TDVPFlexRNNCell_68813966016992
// MI455X (gfx1250) — compile-verified
//
#include <hip/hip_runtime.h>
#include <hip/hip_bf16.h>
#include <stdint.h>

typedef __attribute__((ext_vector_type(2))) float v2f;
typedef __attribute__((ext_vector_type(8))) float v8f;

#define NBATCH 32
#define NSITE  32
#define DB     32          // bond dim
#define PDM    2           // physical dim
#define MW     6           // Hamiltonian MPO bond dim
#define OW     3           // observable MPO bond dim
#define NOBS   2
#define DTV    0.02f
#define NTAY   8

#define SITE_F (DB*PDM*DB)        // 2048 elements per site plane
#define ENVP   (DB*MW*DB)         // 6144 per env plane (dw=6)
#define ENV_F  (2*ENVP)
#define OENVP  (DB*OW*DB)         // 3072 per env plane (dw=3)

// ---- per-batch workspace layout (floats) in d_ws ----
#define RENV_OFF 0
#define RENV_F   (NSITE*ENV_F)            // all right environments
#define LENV_OFF (RENV_OFF + RENV_F)
#define LNEW_OFF (LENV_OFF + ENV_F)
#define TMP1_OFF (LNEW_OFF + ENV_F)
#define TMP1_F   32768
#define TMP2_OFF (TMP1_OFF + TMP1_F)
#define TMP2_F   32768
#define EENV_OFF (TMP2_OFF + TMP2_F)
#define EENV_F   (2*(2*OENVP))
#define ENRM_OFF (EENV_OFF + EENV_F)
#define ENRM_F   (2*(2*DB*DB))
#define WS_PER_B (ENRM_OFF + ENRM_F)      // ~2 MB/batch, L2-resident

#define SMEM_FLOATS 45072

// ---------------------------------------------------------------------------
// CDNA5 async global->LDS staging (ASYNCcnt path).  Each lane moves 16B.
// ---------------------------------------------------------------------------
__device__ __forceinline__ void async_tile_load(float* lds_dst, const float* gsrc,
                                                int nfloats) {
  for (int off = threadIdx.x * 4; off < nfloats; off += blockDim.x * 4) {
    uint32_t laddr = (uint32_t)(uintptr_t)(lds_dst + off);
    const float* g = gsrc + off;
    asm volatile("global_load_async_to_lds_b128 %0, %1, off"
                 :: "v"(laddr), "v"(g)
                 : "memory");
  }
}
__device__ __forceinline__ void wait_async_all() {
  asm volatile("s_wait_asynccnt 0x0" ::: "memory");
}

// ---------------------------------------------------------------------------
// Block-cooperative batched complex GEMM on V_WMMA_F32_16X16X4_F32 (wave32).
//   For bidx in [0,NBT): C_bidx[m,n] (+)= sum_k A_bidx(m,k)*B_bidx(k,n)
// All dims/strides compile-time: no bounds guards, fully unrolled K loop.
// One 16x16 tile per wave; 4 f32 WMMA accumulator chains form the complex MAC.
// Fragment layouts per CDNA5 ISA 7.12.2.
// ---------------------------------------------------------------------------
template<int M, int N, int Kd, int sAm, int sAk, int sBk, int sBn, int sCm, int sCn,
         int NBT = 1, int bA = 0, int bB = 0, int bC = 0>
__device__ __forceinline__ void cgemm(const float* __restrict__ Ar, const float* __restrict__ Ai, float cA,
                                      const float* __restrict__ Br, const float* __restrict__ Bi, float cB,
                                      float* __restrict__ Cr, float* __restrict__ Ci, bool acc)
{
  static_assert(M % 16 == 0 && N % 16 == 0 && Kd % 4 == 0, "tile shapes");
  const int lane = threadIdx.x & 31;
  const int wave = threadIdx.x >> 5;
  const int nw   = blockDim.x >> 5;
  const int row  = lane & 15;
  const int hi   = lane >> 4;
  constexpr int mt = M >> 4, nt = N >> 4, tpb = mt * nt;
  for (int tile = wave; tile < NBT * tpb; tile += nw) {
    const int bidx = tile / tpb;
    const int trem = tile % tpb;
    const int m0 = (trem / nt) << 4;
    const int n0 = (trem % nt) << 4;
    const float* Apr = Ar + bidx * bA + (m0 + row) * sAm;
    const float* Api = Ai + bidx * bA + (m0 + row) * sAm;
    const float* Bpr = Br + bidx * bB + (n0 + row) * sBn;
    const float* Bpi = Bi + bidx * bB + (n0 + row) * sBn;
    v8f rr = {}, ii = {}, ri = {}, ir = {};
#pragma unroll
    for (int k0 = 0; k0 < Kd; k0 += 4) {
      const int ka = k0 + 2 * hi;          // A/B: lanes16-31 hold K+2,K+3
      v2f ar, ai, br, bi;
      ar.x = Apr[ka * sAk];        ar.y = Apr[(ka + 1) * sAk];
      ai.x = cA * Api[ka * sAk];   ai.y = cA * Api[(ka + 1) * sAk];
      br.x = Bpr[ka * sBk];        br.y = Bpr[(ka + 1) * sBk];
      bi.x = cB * Bpi[ka * sBk];   bi.y = cB * Bpi[(ka + 1) * sBk];
      rr = __builtin_amdgcn_wmma_f32_16x16x4_f32(false, ar, false, br, (short)0, rr, false, false);
      ii = __builtin_amdgcn_wmma_f32_16x16x4_f32(false, ai, false, bi, (short)0, ii, false, false);
      ri = __builtin_amdgcn_wmma_f32_16x16x4_f32(false, ar, false, bi, (short)0, ri, false, false);
      ir = __builtin_amdgcn_wmma_f32_16x16x4_f32(false, ai, false, br, (short)0, ir, false, false);
    }
    float* Cpr = Cr + bidx * bC;
    float* Cpi = Ci + bidx * bC;
#pragma unroll
    for (int r = 0; r < 8; ++r) {          // C/D: VGPR r -> rows r / r+8
      const int m = m0 + r + hi * 8;
      const int n = n0 + row;
      const float cre = rr[r] - ii[r];
      const float cim = ri[r] + ir[r];
      if (acc) { Cpr[m*sCm + n*sCn] += cre; Cpi[m*sCm + n*sCn] += cim; }
      else     { Cpr[m*sCm + n*sCn]  = cre; Cpi[m*sCm + n*sCn]  = cim; }
    }
  }
}

__device__ __forceinline__ void zero2(float* a, float* b, int n) {
  for (int e = threadIdx.x; e < n; e += blockDim.x) { a[e] = 0.f; b[e] = 0.f; }
}
__device__ __forceinline__ void copy2(float* dr, float* di, const float* sr, const float* si, int n) {
  for (int e = threadIdx.x; e < n; e += blockDim.x) { dr[e] = sr[e]; di[e] = si[e]; }
}

// ---------------------------------------------------------------------------
// renv_step: Rn[l,w,m] = sum A[l,s,r] R[r,v,n] W[w,p,s,v] conj(A[m,p,n])
// ---------------------------------------------------------------------------
__device__ void renv_step(const float* Ar_, const float* Ai_,   // [l,s,r]
                          const float* Wr, const float* Wi,     // [w,p,s,v] dw=MW
                          const float* Rr, const float* Ri,     // [r,v,n]
                          float* Onr, float* Oni,               // [l,w,m]
                          float* tmp1, float* tmp2)
{
  // t_v[(l,s),n] = A2d @ R_v  (batched over v: 48 tiles)
  cgemm<DB*PDM, DB, DB, DB,1, MW*DB,1, DB,1, MW, 0, DB, 4096>(
      Ar_, Ai_, 1.f, Rr, Ri, 1.f, tmp1, tmp1 + 2048, false);
  __syncthreads();
  // t2_{w,p}[l,n] = sum_{s,v} W[w,p,s,v] * t_v[(l,s),n]
  for (int idx = threadIdx.x; idx < MW*PDM*1024; idx += blockDim.x) {
    const int wp = idx >> 10, e = idx & 1023;
    const int w = wp / PDM, p = wp % PDM;
    const int l = e >> 5, n = e & 31;
    float ar = 0.f, ai = 0.f;
    for (int s = 0; s < PDM; ++s)
      for (int v = 0; v < MW; ++v) {
        const int wi = ((w*PDM + p)*PDM + s)*MW + v;
        const float* t = tmp1 + v * 4096;
        const float xr = t[(l*PDM + s)*DB + n], xi = t[2048 + (l*PDM + s)*DB + n];
        const float wr_ = Wr[wi], wi_ = Wi[wi];
        ar += wr_*xr - wi_*xi; ai += wr_*xi + wi_*xr;
      }
    float* t2 = tmp2 + wp * 2048;
    t2[e] = ar; t2[1024 + e] = ai;
  }
  __syncthreads();
  // Rn_w[l,m] = sum_p t2_{w,p} @ conj(A_p)^T  (batched over w, serial over p)
  for (int p = 0; p < PDM; ++p) {
    cgemm<DB, DB, DB, DB,1, 1,PDM*DB, MW*DB,1, MW, PDM*2048, 0, DB>(
        tmp2 + p*2048, tmp2 + p*2048 + 1024, 1.f,
        Ar_ + p*DB, Ai_ + p*DB, -1.f, Onr, Oni, p != 0);
    __syncthreads();
  }
}

// ---------------------------------------------------------------------------
// env_step (dw templated): Ln[r,v,n] = sum L[a,w,A] A[a,s,r] W[w,p,s,v] conj(A[A,p,n])
// ---------------------------------------------------------------------------
template<int dw>
__device__ void env_step(const float* Lr, const float* Li,      // [a,w,A], a-stride dw*DB
                         const float* Ar_, const float* Ai_,    // ket site [a,s,r]
                         const float* Wr, const float* Wi,      // [w,p,s,v]
                         float* Onr, float* Oni,                // [r,v,n]
                         float* tmp1, float* tmp2)
{
  // t1_w[A,(s,r)] = L_w^T @ A2d  (batched over w)
  cgemm<DB, PDM*DB, DB, 1,dw*DB, PDM*DB,1, PDM*DB,1, dw, DB, 0, 4096>(
      Lr, Li, 1.f, Ar_, Ai_, 1.f, tmp1, tmp1 + 2048, false);
  __syncthreads();
  // t2_{p,v}[A,r] = sum_{w,s} W * t1_w
  for (int idx = threadIdx.x; idx < PDM*dw*1024; idx += blockDim.x) {
    const int pv = idx >> 10, e = idx & 1023;
    const int p = pv / dw, v = pv % dw;
    const int Aq = e >> 5, r = e & 31;
    float ar = 0.f, ai = 0.f;
    for (int w = 0; w < dw; ++w)
      for (int s = 0; s < PDM; ++s) {
        const int wi = ((w*PDM + p)*PDM + s)*dw + v;
        const float* t1 = tmp1 + w * 4096;
        const float xr = t1[Aq*64 + s*DB + r], xi = t1[2048 + Aq*64 + s*DB + r];
        const float wr_ = Wr[wi], wi_ = Wi[wi];
        ar += wr_*xr - wi_*xi; ai += wr_*xi + wi_*xr;
      }
    float* t2 = tmp2 + pv * 2048;
    t2[e] = ar; t2[1024 + e] = ai;
  }
  __syncthreads();
  // Ln_v[r,n] = sum_p t2_{p,v}^T @ conj(A_p)  (batched over v, serial over p)
  for (int p = 0; p < PDM; ++p) {
    cgemm<DB, DB, DB, 1,DB, PDM*DB,1, dw*DB,1, dw, 2048, 0, DB>(
        tmp2 + p*dw*2048, tmp2 + p*dw*2048 + 1024, 1.f,
        Ar_ + p*DB, Ai_ + p*DB, -1.f, Onr, Oni, p != 0);
    __syncthreads();
  }
}

// ---------------------------------------------------------------------------
// heff_mv: out[A,p,B] = sum L[a,w,A] W[w,p,s,v] R[b,v,B] x[a,s,b]
// ---------------------------------------------------------------------------
__device__ void heff_mv(const float* Lr, const float* Li,
                        const float* Wr, const float* Wi,
                        const float* Rr, const float* Ri,
                        const float* xr, const float* xi,
                        float* outr, float* outi,
                        float* Yw, float* Tvp)
{
  zero2(outr, outi, SITE_F);
  __syncthreads();
  for (int s = 0; s < PDM; ++s) {
    // Y_w = L_w^T @ x_s for all w (batched over w)
    cgemm<DB, DB, DB, 1,MW*DB, PDM*DB,1, DB,1, MW, DB, 0, 2048>(
        Lr, Li, 1.f, xr + s*DB, xi + s*DB, 1.f, Yw, Yw + 1024, false);
    __syncthreads();
    for (int v = 0; v < MW; ++v) {
      // T_p = sum_w W[w,p,s,v] * Y_w   (both p at once)
      for (int e = threadIdx.x; e < 2048; e += blockDim.x) {
        const int p = e >> 10, e1 = e & 1023;
        float ar = 0.f, ai = 0.f;
        for (int w = 0; w < MW; ++w) {
          const int wi = ((w*PDM + p)*PDM + s)*MW + v;
          const float wr_ = Wr[wi], wi_ = Wi[wi];
          const float yr_ = Yw[w*2048 + e1], yi_ = Yw[w*2048 + 1024 + e1];
          ar += wr_*yr_ - wi_*yi_; ai += wr_*yi_ + wi_*yr_;
        }
        Tvp[p*2048 + e1] = ar; Tvp[p*2048 + 1024 + e1] = ai;
      }
      __syncthreads();
      // out[:,p,:] += T_p @ R_v  (batched over p)
      cgemm<DB, DB, DB, DB,1, MW*DB,1, PDM*DB,1, PDM, 2048, 0, DB>(
          Tvp, Tvp + 1024, 1.f, Rr + v*DB, Ri + v*DB, 1.f, outr, outi, true);
      __syncthreads();
    }
  }
}

// keff_mv: out[A,B] = sum_w (L_w^T @ x) @ R_w   (T scratch: MW slots of 2048)
__device__ void keff_mv(const float* Lr, const float* Li,
                        const float* Rr, const float* Ri,
                        const float* xr, const float* xi,
                        float* outr, float* outi, float* T)
{
  // T_w = L_w^T @ x for all w (batched over w)
  cgemm<DB, DB, DB, 1,MW*DB, DB,1, DB,1, MW, DB, 0, 2048>(
      Lr, Li, 1.f, xr, xi, 1.f, T, T + 1024, false);
  __syncthreads();
  for (int w = 0; w < MW; ++w) {
    cgemm<DB, DB, DB, DB,1, MW*DB,1, DB,1>(
        T + w*2048, T + w*2048 + 1024, 1.f,
        Rr + w*DB, Ri + w*DB, 1.f, outr, outi, w != 0);
    __syncthreads();
  }
}

// ---------------------------------------------------------------------------
// Full TDVP sweep: one 8-wave workgroup per batch chain.
// ---------------------------------------------------------------------------
__global__ __launch_bounds__(256, 1)
void tdvp_sweep_kernel(const float* __restrict__ mps_re, const float* __restrict__ mps_im,
                       const float* __restrict__ mpo_re, const float* __restrict__ mpo_im,
                       const float* __restrict__ ops_re, const float* __restrict__ ops_im,
                       float* __restrict__ out, float* __restrict__ ws)
{
  extern __shared__ float sm[];
  float* yr  = sm;                   // evolved site tensor / Q  (2048 re)
  float* yi  = yr  + SITE_F;
  float* txr = yi  + SITE_F;         // Taylor term
  float* txi = txr + SITE_F;
  float* tnr = txi + SITE_F;         // H_eff output
  float* tni = tnr + SITE_F;
  float* Yw  = tni + SITE_F;         // MW partial GEMMs (6 x 1024 cplx)
  float* Tvp = Yw  + MW*2048;        // 2 x 1024 cplx scratch
  float* Cbr = Tvp + 4096;  float* Cbi = Cbr + 1024;   // bond tensor C
  float* Ktr = Cbi + 1024;  float* Kti = Ktr + 1024;   // bond Taylor term
  float* Knr = Kti + 1024;  float* Kni = Knr + 1024;   // K_eff output
  float* Qr  = Kni + 1024;  float* Qi  = Qr  + 1024;   // R from QR
  float* red = Qi  + 1024;           // small reduction slots (16)
  float* Sg  = red + 16;             // async double-buffer: 2 x (re2048 + im2048)

  const int b = blockIdx.x;
  float* wsb = ws + (size_t)b * WS_PER_B;
  const float* tsr = mps_re + (size_t)b * NSITE * SITE_F;
  const float* tsi = mps_im + (size_t)b * NSITE * SITE_F;
  float* outRe = out + NBATCH * NOBS;
  float* outIm = outRe + (size_t)NBATCH * NSITE * SITE_F;
  float* qbr = outRe + (size_t)b * NSITE * SITE_F;
  float* qbi = outIm + (size_t)b * NSITE * SITE_F;

  float* renv = wsb + RENV_OFF;
  float* tmp1 = wsb + TMP1_OFF;
  float* tmp2 = wsb + TMP2_OFF;

  // ---------------- Phase A: right environments (stored per site in L2) ----
  {
    float* Rr = renv + (NSITE - 1) * ENV_F;
    float* Ri = Rr + ENVP;
    for (int e = threadIdx.x; e < ENVP; e += blockDim.x) {
      const int r = e / (MW*DB), v = (e / DB) % MW, n = e % DB;
      Rr[e] = (v == MW - 1 && r == n) ? 1.f : 0.f;
      Ri[e] = 0.f;
    }
    __syncthreads();
  }
  for (int i = NSITE - 1; i >= 1; --i) {
    if (i > 1) { __builtin_prefetch(tsr + (i-1)*SITE_F, 0, 1);
                 __builtin_prefetch(tsi + (i-1)*SITE_F, 0, 1); }
    const float* Rr = renv + i * ENV_F;
    renv_step(tsr + i*SITE_F, tsi + i*SITE_F,
              mpo_re + i*144, mpo_im + i*144,
              Rr, Rr + ENVP,
              renv + (i-1)*ENV_F, renv + (i-1)*ENV_F + ENVP,
              tmp1, tmp2);
  }

  // ---------------- Phase B: left-to-right one-site TDVP sweep -------------
  float* Lcur_r = wsb + LENV_OFF; float* Lcur_i = Lcur_r + ENVP;
  float* Lalt_r = wsb + LNEW_OFF; float* Lalt_i = Lalt_r + ENVP;
  for (int e = threadIdx.x; e < ENVP; e += blockDim.x) {
    const int a = e / (MW*DB), w = (e / DB) % MW, A = e % DB;
    Lcur_r[e] = (w == 0 && a == A) ? 1.f : 0.f;
    Lcur_i[e] = 0.f;
  }
  for (int e = threadIdx.x; e < DB*DB; e += blockDim.x) {
    Cbr[e] = ((e >> 5) == (e & 31)) ? 1.f : 0.f;
    Cbi[e] = 0.f;
  }
  // stage site 0 into LDS through the async path
  async_tile_load(Sg + 0,      tsr, SITE_F);
  async_tile_load(Sg + SITE_F, tsi, SITE_F);
  int cur = 0;
  __syncthreads();

  for (int i = 0; i < NSITE; ++i) {
    const float* Rr = renv + i * ENV_F;   const float* Ri = Rr + ENVP;
    const float* Wsr = mpo_re + i * 144;  const float* Wsi = mpo_im + i * 144;
    float* SgR = Sg + cur * (2*SITE_F);
    float* SgI = SgR + SITE_F;

    wait_async_all();       // own wave's async loads done
    __syncthreads();        // all waves' loads visible

    // absorb bond tensor: y[a,(s,r)] = C @ ts_i   (ts_i staged in LDS)
    cgemm<DB, PDM*DB, DB, DB,1, PDM*DB,1, PDM*DB,1>(
        Cbr, Cbi, 1.f, SgR, SgI, 1.f, yr, yi, false);
    __syncthreads();

    // kick off next site's staging; overlaps with the Taylor evolution below
    if (i + 1 < NSITE) {
      float* SgRn = Sg + (cur ^ 1) * (2*SITE_F);
      async_tile_load(SgRn,          tsr + (i+1)*SITE_F, SITE_F);
      async_tile_load(SgRn + SITE_F, tsi + (i+1)*SITE_F, SITE_F);
    }

    copy2(txr, txi, yr, yi, SITE_F);
    __syncthreads();

    // y = exp(-i dt Heff) x  (order-8 Taylor of WMMA matvecs)
    for (int k = 1; k <= NTAY; ++k) {
      heff_mv(Lcur_r, Lcur_i, Wsr, Wsi, Rr, Ri, txr, txi, tnr, tni, Yw, Tvp);
      const float cik = -DTV / (float)k;      // coef = -i*dt/k
      for (int e = threadIdx.x; e < SITE_F; e += blockDim.x) {
        const float ar = tnr[e], ai = tni[e];
        const float nr_ = -cik * ai, ni_ = cik * ar;
        txr[e] = nr_; txi[e] = ni_;
        yr[e] += nr_; yi[e] += ni_;
      }
      __syncthreads();
    }

    if (i < NSITE - 1) {
      // ---- QR (modified Gram-Schmidt, 64x32, wave32 shfl butterflies) ----
      zero2(Qr, Qi, DB*DB);
      __syncthreads();
      {
        const int lane = threadIdx.x & 31;
        const int wv   = threadIdx.x >> 5;
        for (int j = 0; j < DB; ++j) {
          float a0r = yr[lane*DB + j],      a0i = yi[lane*DB + j];
          float a1r = yr[(lane+32)*DB + j], a1i = yi[(lane+32)*DB + j];
          float ss = a0r*a0r + a0i*a0i + a1r*a1r + a1i*a1i;
          for (int off = 16; off > 0; off >>= 1) ss += __shfl_xor(ss, off, 32);
          const float nrm = sqrtf(ss);
          const float inv = 1.0f / (nrm + 1e-30f);
          if (wv == 0) {
            yr[lane*DB + j] = a0r*inv;      yi[lane*DB + j] = a0i*inv;
            yr[(lane+32)*DB + j] = a1r*inv; yi[(lane+32)*DB + j] = a1i*inv;
            if (lane == 0) Qr[j*DB + j] = nrm;
          }
          __syncthreads();
          for (int kc = j + 1; kc < DB; ++kc) {
            if ((kc & 7) != wv) continue;   // wave-uniform column ownership
            const float q0r = yr[lane*DB + j], q0i = yi[lane*DB + j];
            const float q1r = yr[(lane+32)*DB + j], q1i = yi[(lane+32)*DB + j];
            float x0r = yr[lane*DB + kc], x0i = yi[lane*DB + kc];
            float x1r = yr[(lane+32)*DB + kc], x1i = yi[(lane+32)*DB + kc];
            float pr = q0r*x0r + q0i*x0i + q1r*x1r + q1i*x1i;   // q^H x
            float pi = q0r*x0i - q0i*x0r + q1r*x1i - q1i*x1r;
            for (int off = 16; off > 0; off >>= 1) {
              pr += __shfl_xor(pr, off, 32); pi += __shfl_xor(pi, off, 32);
            }
            x0r -= pr*q0r - pi*q0i;  x0i -= pr*q0i + pi*q0r;
            x1r -= pr*q1r - pi*q1i;  x1i -= pr*q1i + pi*q1r;
            yr[lane*DB + kc] = x0r; yi[lane*DB + kc] = x0i;
            yr[(lane+32)*DB + kc] = x1r; yi[(lane+32)*DB + kc] = x1i;
            if (lane == 0) { Qr[j*DB + kc] = pr; Qi[j*DB + kc] = pi; }
          }
          __syncthreads();
        }
      }

      // write Q as the updated site tensor
      for (int e = threadIdx.x; e < SITE_F; e += blockDim.x) {
        qbr[(size_t)i*SITE_F + e] = yr[e];
        qbi[(size_t)i*SITE_F + e] = yi[e];
      }
      __syncthreads();

      // advance left environment with Q
      env_step<MW>(Lcur_r, Lcur_i, yr, yi, Wsr, Wsi, Lalt_r, Lalt_i, tmp1, tmp2);
      { float* t;
        t = Lcur_r; Lcur_r = Lalt_r; Lalt_r = t;
        t = Lcur_i; Lcur_i = Lalt_i; Lalt_i = t; }

      // backward bond step: C = exp(+i dt Keff) Rm
      copy2(Cbr, Cbi, Qr, Qi, DB*DB);
      copy2(Ktr, Kti, Qr, Qi, DB*DB);
      __syncthreads();
      for (int k = 1; k <= NTAY; ++k) {
        keff_mv(Lcur_r, Lcur_i, Rr, Ri, Ktr, Kti, Knr, Kni, Yw);
        const float cik = DTV / (float)k;     // coef = +i*dt/k
        for (int e = threadIdx.x; e < DB*DB; e += blockDim.x) {
          const float ar = Knr[e], ai = Kni[e];
          const float nr_ = -cik * ai, ni_ = cik * ar;
          Ktr[e] = nr_; Kti[e] = ni_;
          Cbr[e] += nr_; Cbi[e] += ni_;
        }
        __syncthreads();
      }
    } else {
      // last site: evolved (non-orthonormalized) tensor is the output
      for (int e = threadIdx.x; e < SITE_F; e += blockDim.x) {
        qbr[(size_t)i*SITE_F + e] = yr[e];
        qbi[(size_t)i*SITE_F + e] = yi[e];
      }
      __syncthreads();
    }
    cur ^= 1;
  }

  __threadfence();
  __syncthreads();

  // ---------------- Phase C: norm + observable expectation values ----------
  float* E0r = wsb + ENRM_OFF;  float* E0i = E0r + 1024;
  float* E1r = E0r + 2048;      float* E1i = E1r + 1024;
  for (int e = threadIdx.x; e < DB*DB; e += blockDim.x) {
    E0r[e] = ((e >> 5) == (e & 31)) ? 1.f : 0.f;  E0i[e] = 0.f;
  }
  __syncthreads();
  for (int i = 0; i < NSITE; ++i) {
    const float* qsr = qbr + (size_t)i * SITE_F;
    const float* qsi = qbi + (size_t)i * SITE_F;
    cgemm<DB, PDM*DB, DB, 1,DB, PDM*DB,1, PDM*DB,1>(
        E0r, E0i, 1.f, qsr, qsi, 1.f, tmp1, tmp1 + 2048, false);
    __syncthreads();
    cgemm<DB, DB, PDM*DB, 1,DB, DB,1, DB,1>(
        tmp1, tmp1 + 2048, 1.f, qsr, qsi, -1.f, E1r, E1i, false);
    __syncthreads();
    { float* t;
      t = E0r; E0r = E1r; E1r = t;
      t = E0i; E0i = E1i; E1i = t; }
  }
  if (threadIdx.x < 32) {
    float trr = E0r[threadIdx.x * 33], tri = E0i[threadIdx.x * 33];
    for (int off = 16; off > 0; off >>= 1) {
      trr += __shfl_xor(trr, off, 32); tri += __shfl_xor(tri, off, 32);
    }
    if (threadIdx.x == 0) { red[0] = trr; red[1] = tri; }
  }
  __syncthreads();

  for (int kb = 0; kb < NOBS; ++kb) {
    float* G0r = wsb + EENV_OFF;    float* G0i = G0r + OENVP;
    float* G1r = G0r + 2 * OENVP;   float* G1i = G1r + OENVP;
    for (int e = threadIdx.x; e < OENVP; e += blockDim.x) {
      const int a = e / (OW*DB), w = (e / DB) % OW, A = e % DB;
      G0r[e] = (w == 0 && a == A) ? 1.f : 0.f;
      G0i[e] = 0.f;
    }
    __syncthreads();
    for (int i = 0; i < NSITE; ++i) {
      env_step<OW>(G0r, G0i, qbr + (size_t)i*SITE_F, qbi + (size_t)i*SITE_F,
                   ops_re + ((size_t)kb*NSITE + i)*36, ops_im + ((size_t)kb*NSITE + i)*36,
                   G1r, G1i, tmp1, tmp2);
      { float* t;
        t = G0r; G0r = G1r; G1r = t;
        t = G0i; G0i = G1i; G1i = t; }
    }
    if (threadIdx.x < 32) {
      const int a = threadIdx.x;
      float evr = G0r[a*(OW*DB) + (OW-1)*DB + a];
      float evi = G0i[a*(OW*DB) + (OW-1)*DB + a];
      for (int off = 16; off > 0; off >>= 1) {
        evr += __shfl_xor(evr, off, 32); evi += __shfl_xor(evi, off, 32);
      }
      if (threadIdx.x == 0) {
        const float nr = red[0], ni = red[1];
        out[b*NOBS + kb] = (evr*nr + evi*ni) / (nr*nr + ni*ni + 1e-30f);
      }
    }
    __syncthreads();
  }
}

extern "C" void kernel_launch(void* const* d_in, const int* in_sizes, int n_in,
                              void* d_out, int out_size, void* d_ws, size_t ws_size,
                              hipStream_t stream) {
  (void)in_sizes; (void)n_in; (void)out_size; (void)ws_size;
  const float* mps_re = (const float*)d_in[0];
  const float* mps_im = (const float*)d_in[1];
  const float* mpo_re = (const float*)d_in[2];
  const float* mpo_im = (const float*)d_in[3];
  const float* ops_re = (const float*)d_in[4];
  const float* ops_im = (const float*)d_in[5];
  dim3 grid(NBATCH), block(256);
  const size_t smem = (size_t)SMEM_FLOATS * sizeof(float);
  tdvp_sweep_kernel<<<grid, block, smem, stream>>>(
      mps_re, mps_im, mpo_re, mpo_im, ops_re, ops_im,
      (float*)d_out, (float*)d_ws);
}